// Gemma3MonolithicWrapper_9861244911652
// MI455X (gfx1250) — compile-verified
//
#include <hip/hip_runtime.h>
#include <hip/hip_bf16.h>
#include <math.h>

#define HID   1152
#define NH    4
#define DH    256
#define FF    6912
#define VOCAB 64000
#define SEQ   4096
#define NL    12

typedef __attribute__((ext_vector_type(2))) float v2f;
typedef __attribute__((ext_vector_type(8))) float v8f;
typedef int b128_t __attribute__((vector_size(16)));   // matches builtin's V4i

#if defined(__has_builtin)
#if __has_builtin(__builtin_amdgcn_wmma_f32_16x16x4_f32)
#define USE_WMMA 1
#endif
#if __has_builtin(__builtin_amdgcn_global_load_async_to_lds_b128) && \
    __has_builtin(__builtin_amdgcn_s_wait_asynccnt)
#define USE_ASYNC 1
#endif
#endif
#ifndef USE_WMMA
#define USE_WMMA 0
#endif
#ifndef USE_ASYNC
#define USE_ASYNC 0
#endif

#define GAS1(p) ((__attribute__((address_space(1))) b128_t*)(p))
#define LAS3(p) ((__attribute__((address_space(3))) b128_t*)(p))

// ---------------- embed lookup + scale ----------------
__global__ void k_embed(const int* __restrict__ ids, const float* __restrict__ embed,
                        float* __restrict__ h) {
  int i = blockIdx.x * blockDim.x + threadIdx.x;
  if (i < HID) h[i] = embed[(size_t)ids[0] * HID + i] * 33.941125497f; // sqrt(1152)
}

// ---------------- rmsnorm: y = x * rsqrt(mean(x^2)+eps) * w ----------------
__global__ __launch_bounds__(256) void k_rms(const float* __restrict__ x,
                                             const float* __restrict__ w,
                                             float* __restrict__ y, int n) {
  __shared__ float red[8];
  float ss = 0.f;
  for (int i = threadIdx.x; i < n; i += 256) { float v = x[i]; ss += v * v; }
  for (int off = 16; off; off >>= 1) ss += __shfl_xor(ss, off, 32);
  if ((threadIdx.x & 31) == 0) red[threadIdx.x >> 5] = ss;
  __syncthreads();
  float tot = 0.f;
#pragma unroll
  for (int i = 0; i < 8; ++i) tot += red[i];
  float r = rsqrtf(tot / (float)n + 1e-6f);
  for (int i = threadIdx.x; i < n; i += 256) y[i] = x[i] * r * w[i];
}

// ---------------- residual add of rmsnorm: h += rms(t)*w ----------------
__global__ __launch_bounds__(256) void k_add_rms(float* __restrict__ h,
                                                 const float* __restrict__ t,
                                                 const float* __restrict__ w, int n) {
  __shared__ float red[8];
  float ss = 0.f;
  for (int i = threadIdx.x; i < n; i += 256) { float v = t[i]; ss += v * v; }
  for (int off = 16; off; off >>= 1) ss += __shfl_xor(ss, off, 32);
  if ((threadIdx.x & 31) == 0) red[threadIdx.x >> 5] = ss;
  __syncthreads();
  float tot = 0.f;
#pragma unroll
  for (int i = 0; i < 8; ++i) tot += red[i];
  float r = rsqrtf(tot / (float)n + 1e-6f);
  for (int i = threadIdx.x; i < n; i += 256) h[i] += t[i] * r * w[i];
}

// ---------------- GEMV y[M] = W[M,K] @ x[K], 16 rows/block, WMMA f32 16x16x4 ------
// K multiple of 128 (template param). grid.x = M/16, block = 256 (8 waves).
// Double-buffered LDS tiles filled by async global->LDS copies (ASYNCcnt),
// overlapped with WMMA compute on the other buffer.
template <int K>
__global__ __launch_bounds__(256) void k_gemv(const float* __restrict__ W,
                                              const float* __restrict__ x,
                                              float* __restrict__ y) {
  __shared__ float xs[K];               // whole x vector
  __shared__ float tile[2][16 * 132];   // double-buffered 16x128 tile, stride 132
  __shared__ float part[128];           // 8 waves x 16 rows partials
  const int tid = threadIdx.x;
  const int wave = tid >> 5, lane = tid & 31;
  const int row0 = blockIdx.x * 16;
  const int r = wave * 2;

  for (int i = tid; i < K; i += 256) xs[i] = x[i];

  const int m  = lane & 15;
  const int hi = (lane & 16) ? 1 : 0;
  const int k0 = hi ? 2 : 0, k1 = hi ? 3 : 1;

#if USE_WMMA
  v8f acc = {};
#else
  float facc = 0.f;
#endif

  const int nch = K >> 7;

#if USE_ASYNC
  // prologue: async-load chunk 0 into buffer 0
  __builtin_amdgcn_global_load_async_to_lds_b128(
      GAS1(W + (size_t)(row0 + r) * K + lane * 4),
      LAS3(&tile[0][r * 132 + lane * 4]), 0, 0);
  __builtin_amdgcn_global_load_async_to_lds_b128(
      GAS1(W + (size_t)(row0 + r + 1) * K + lane * 4),
      LAS3(&tile[0][(r + 1) * 132 + lane * 4]), 0, 0);
#endif

  for (int c = 0; c < nch; ++c) {
    const int cb = c << 7;
    const int buf = c & 1;
#if USE_ASYNC
    if (c + 1 < nch) {
      const int nb = cb + 128;
      __builtin_amdgcn_global_load_async_to_lds_b128(
          GAS1(W + (size_t)(row0 + r) * K + nb + lane * 4),
          LAS3(&tile[buf ^ 1][r * 132 + lane * 4]), 0, 0);
      __builtin_amdgcn_global_load_async_to_lds_b128(
          GAS1(W + (size_t)(row0 + r + 1) * K + nb + lane * 4),
          LAS3(&tile[buf ^ 1][(r + 1) * 132 + lane * 4]), 0, 0);
      __builtin_amdgcn_s_wait_asynccnt(2);  // chunk c complete, c+1 in flight
    } else {
      __builtin_amdgcn_s_wait_asynccnt(0);
    }
    __syncthreads();
#else
    __syncthreads();
    {
      const float4 t0 = *(const float4*)(W + (size_t)(row0 + r)     * K + cb + lane * 4);
      const float4 t1 = *(const float4*)(W + (size_t)(row0 + r + 1) * K + cb + lane * 4);
      float* tp0 = &tile[buf][r * 132 + lane * 4];
      tp0[0] = t0.x; tp0[1] = t0.y; tp0[2] = t0.z; tp0[3] = t0.w;
      float* tp1 = &tile[buf][(r + 1) * 132 + lane * 4];
      tp1[0] = t1.x; tp1[1] = t1.y; tp1[2] = t1.z; tp1[3] = t1.w;
    }
    __syncthreads();
#endif
    const int kw = wave * 16;
#if USE_WMMA
#pragma unroll
    for (int kk = 0; kk < 16; kk += 4) {
      const int kb = kw + kk;
      v2f a, b;
      a.x = tile[buf][m * 132 + kb + k0];
      a.y = tile[buf][m * 132 + kb + k1];
      b.x = xs[cb + kb + k0];          // x replicated over all 16 N columns
      b.y = xs[cb + kb + k1];
      acc = __builtin_amdgcn_wmma_f32_16x16x4_f32(false, a, false, b,
                                                  (short)0, acc, false, false);
    }
#else
    if (lane < 16) {
#pragma unroll
      for (int kk = 0; kk < 16; ++kk)
        facc += tile[buf][m * 132 + kw + kk] * xs[cb + kw + kk];
    }
#endif
    __syncthreads();   // all waves done with tile[buf] before it is refilled
  }

#if USE_WMMA
  // C layout: VGPR i, lane0 = (M=i, N=0); lane16 = (M=i+8, N=0)
  if (lane == 0) {
#pragma unroll
    for (int i = 0; i < 8; ++i) part[wave * 16 + i] = acc[i];
  }
  if (lane == 16) {
#pragma unroll
    for (int i = 0; i < 8; ++i) part[wave * 16 + 8 + i] = acc[i];
  }
#else
  if (lane < 16) part[wave * 16 + m] = facc;
#endif
  __syncthreads();
  if (tid < 16) {
    float s = 0.f;
#pragma unroll
    for (int w = 0; w < 8; ++w) s += part[w * 16 + tid];
    y[row0 + tid] = s;
  }
}

// ---------------- per-head RMSNorm + RoPE for q (4 heads) and k (1 head) --------
__global__ __launch_bounds__(256) void k_qkrope(float* __restrict__ q, float* __restrict__ k,
                                                const float* __restrict__ qnw,
                                                const float* __restrict__ knw,
                                                const float* __restrict__ cost,
                                                const float* __restrict__ sint,
                                                const int* __restrict__ posp) {
  const int wave = threadIdx.x >> 5, lane = threadIdx.x & 31;
  if (wave >= NH + 1) return;
  float* vec       = (wave < NH) ? q + wave * DH : k;
  const float* wn  = (wave < NH) ? qnw : knw;
  const int i0 = lane * 4;
  float lo[4], hif[4];
#pragma unroll
  for (int j = 0; j < 4; ++j) { lo[j] = vec[i0 + j]; hif[j] = vec[i0 + 128 + j]; }
  float ss = 0.f;
#pragma unroll
  for (int j = 0; j < 4; ++j) ss += lo[j] * lo[j] + hif[j] * hif[j];
  for (int off = 16; off; off >>= 1) ss += __shfl_xor(ss, off, 32);
  const float r = rsqrtf(ss / (float)DH + 1e-6f);
  const int p = posp[0];
  const float* cp = cost + (size_t)p * DH;
  const float* sp = sint + (size_t)p * DH;
#pragma unroll
  for (int j = 0; j < 4; ++j) {
    float nl = lo[j]  * r * wn[i0 + j];
    float nh = hif[j] * r * wn[i0 + 128 + j];
    vec[i0 + j]       = nl * cp[i0 + j]       - nh * sp[i0 + j];        // rot_half = [-b, a]
    vec[i0 + 128 + j] = nh * cp[i0 + 128 + j] + nl * sp[i0 + 128 + j];
  }
}

// ---------------- flash attention: 1 query, 4 heads, KV=1, S=4096 ----------------
__global__ __launch_bounds__(256) void k_attn(const float* __restrict__ q,
                                              const float* __restrict__ knew,
                                              const float* __restrict__ vnew,
                                              const float* __restrict__ Kc,
                                              const float* __restrict__ Vc,
                                              const float* __restrict__ mask,
                                              const int* __restrict__ posp,
                                              float* __restrict__ o) {
  __shared__ float qs[NH * DH];
  __shared__ float cm[8][NH];
  __shared__ float csum[8][NH];
  __shared__ float cacc[8][NH][DH];
  const int tid = threadIdx.x, wave = tid >> 5, lane = tid & 31;
  for (int i = tid; i < NH * DH; i += 256) qs[i] = q[i];
  __syncthreads();
  float qreg[NH][8];
#pragma unroll
  for (int h = 0; h < NH; ++h)
#pragma unroll
    for (int j = 0; j < 8; ++j) qreg[h][j] = qs[h * DH + lane * 8 + j];

  float mh[NH], sh[NH], acc[NH][8];
#pragma unroll
  for (int h = 0; h < NH; ++h) {
    mh[h] = -1e30f; sh[h] = 0.f;
#pragma unroll
    for (int j = 0; j < 8; ++j) acc[h][j] = 0.f;
  }
  const int p = posp[0];
  for (int s = wave; s < SEQ; s += 8) {
    const float* kp = (s == p) ? knew : Kc + (size_t)s * DH;
    const float* vp = (s == p) ? vnew : Vc + (size_t)s * DH;
    if (s + 8 < SEQ) {   // prefetch next iteration's rows (global_prefetch_b8)
      __builtin_prefetch(Kc + (size_t)(s + 8) * DH + lane * 8, 0, 0);
      __builtin_prefetch(Vc + (size_t)(s + 8) * DH + lane * 8, 0, 0);
    }
    const float4 ka = ((const float4*)kp)[lane * 2];
    const float4 kb = ((const float4*)kp)[lane * 2 + 1];
    const float4 va = ((const float4*)vp)[lane * 2];
    const float4 vb = ((const float4*)vp)[lane * 2 + 1];
    const float kv[8] = {ka.x, ka.y, ka.z, ka.w, kb.x, kb.y, kb.z, kb.w};
    const float vv[8] = {va.x, va.y, va.z, va.w, vb.x, vb.y, vb.z, vb.w};
    const float msk = mask[s];
#pragma unroll
    for (int h = 0; h < NH; ++h) {
      float d0 = 0.f;
#pragma unroll
      for (int j = 0; j < 8; ++j) d0 += qreg[h][j] * kv[j];
      for (int off = 16; off; off >>= 1) d0 += __shfl_xor(d0, off, 32);
      const float sc = d0 * 0.0625f + msk;              // SCALE = 256^-0.5
      const float nm = fmaxf(mh[h], sc);
      const float f  = __expf(mh[h] - nm);
      const float pr = __expf(sc - nm);
      sh[h] = sh[h] * f + pr;
#pragma unroll
      for (int j = 0; j < 8; ++j) acc[h][j] = acc[h][j] * f + pr * vv[j];
      mh[h] = nm;
    }
  }
  if (lane == 0)
#pragma unroll
    for (int h = 0; h < NH; ++h) { cm[wave][h] = mh[h]; csum[wave][h] = sh[h]; }
#pragma unroll
  for (int h = 0; h < NH; ++h)
#pragma unroll
    for (int j = 0; j < 8; ++j) cacc[wave][h][lane * 8 + j] = acc[h][j];
  __syncthreads();
  for (int i = tid; i < NH * DH; i += 256) {
    const int h = i >> 8, d = i & 255;
    float gm = -1e30f;
#pragma unroll
    for (int w = 0; w < 8; ++w) gm = fmaxf(gm, cm[w][h]);
    float tot = 0.f, val = 0.f;
#pragma unroll
    for (int w = 0; w < 8; ++w) {
      const float f = __expf(cm[w][h] - gm);
      tot += csum[w][h] * f;
      val += cacc[w][h][d] * f;
    }
    o[i] = val / tot;
  }
}

// ---------------- gelu(tanh) gate * up ----------------
__global__ void k_gelu_mul(const float* __restrict__ g, const float* __restrict__ u,
                           float* __restrict__ out, int n) {
  int i = blockIdx.x * blockDim.x + threadIdx.x;
  if (i < n) {
    float x = g[i];
    float t = tanhf(0.7978845608f * (x + 0.044715f * x * x * x));
    out[i] = 0.5f * x * (1.f + t) * u[i];
  }
}

// ---------------- argmax over logits ----------------
__global__ __launch_bounds__(1024) void k_argmax(const float* __restrict__ logits,
                                                 float* __restrict__ out) {
  __shared__ float bm[32];
  __shared__ int bi[32];
  const int tid = threadIdx.x;
  float best = -3.4e38f; int bidx = 0;
  for (int i = tid; i < VOCAB; i += 1024) {
    float v = logits[i];
    if (v > best) { best = v; bidx = i; }
  }
  for (int off = 16; off; off >>= 1) {
    float ov = __shfl_xor(best, off, 32);
    int   oi = __shfl_xor(bidx, off, 32);
    if (ov > best || (ov == best && oi < bidx)) { best = ov; bidx = oi; }
  }
  if ((tid & 31) == 0) { bm[tid >> 5] = best; bi[tid >> 5] = bidx; }
  __syncthreads();
  if (tid == 0) {
    for (int w = 1; w < 32; ++w)
      if (bm[w] > best || (bm[w] == best && bi[w] < bidx)) { best = bm[w]; bidx = bi[w]; }
    out[0] = (float)bidx;
    out[1] = best;
  }
}

extern "C" void kernel_launch(void* const* d_in, const int* in_sizes, int n_in,
                              void* d_out, int out_size, void* d_ws, size_t ws_size,
                              hipStream_t stream) {
  (void)in_sizes; (void)n_in; (void)out_size; (void)ws_size;
  const int*   ids    = (const int*)  d_in[0];
  const int*   pos    = (const int*)  d_in[1];
  const float* cmask  = (const float*)d_in[2];
  // d_in[3] update_mask: superseded by position substitution in k_attn
  const float* embed  = (const float*)d_in[4];
  const float* Wq     = (const float*)d_in[5];
  const float* Wk     = (const float*)d_in[6];
  const float* Wv     = (const float*)d_in[7];
  const float* Wo     = (const float*)d_in[8];
  const float* Wg     = (const float*)d_in[9];
  const float* Wu     = (const float*)d_in[10];
  const float* Wd     = (const float*)d_in[11];
  const float* ln_in  = (const float*)d_in[12];
  const float* ln_pa  = (const float*)d_in[13];
  const float* ln_pf  = (const float*)d_in[14];
  const float* ln_ff  = (const float*)d_in[15];
  const float* qnw    = (const float*)d_in[16];
  const float* knw    = (const float*)d_in[17];
  const float* nw     = (const float*)d_in[18];
  const float* lmh    = (const float*)d_in[19];
  const float* kvc    = (const float*)d_in[20];
  const float* cos_s  = (const float*)d_in[21];
  const float* sin_s  = (const float*)d_in[22];
  const float* cos_f  = (const float*)d_in[23];
  const float* sin_f  = (const float*)d_in[24];

  float* ws = (float*)d_ws;
  float* h      = ws;                 // 1152
  float* xb     = h + HID;            // 1152
  float* q      = xb + HID;           // 1024
  float* kvec   = q + NH * DH;        // 256
  float* vvec   = kvec + DH;          // 256
  float* o      = vvec + DH;          // 1024
  float* t1     = o + NH * DH;        // 1152
  float* gate   = t1 + HID;           // 6912
  float* up     = gate + FF;          // 6912
  float* logits = up + FF;            // 64000

  k_embed<<<(HID + 255) / 256, 256, 0, stream>>>(ids, embed, h);

  for (int l = 0; l < NL; ++l) {
    const bool full = ((l + 1) % 6 == 0);
    const float* ct = full ? cos_f : cos_s;
    const float* st = full ? sin_f : sin_s;

    k_rms<<<1, 256, 0, stream>>>(h, ln_in + l * HID, xb, HID);
    k_gemv<HID><<<NH * DH / 16, 256, 0, stream>>>(Wq + (size_t)l * NH * DH * HID, xb, q);
    k_gemv<HID><<<DH / 16, 256, 0, stream>>>(Wk + (size_t)l * DH * HID, xb, kvec);
    k_gemv<HID><<<DH / 16, 256, 0, stream>>>(Wv + (size_t)l * DH * HID, xb, vvec);
    k_qkrope<<<1, 256, 0, stream>>>(q, kvec, qnw + l * DH, knw + l * DH, ct, st, pos);
    k_attn<<<1, 256, 0, stream>>>(q, kvec, vvec,
                                  kvc + (size_t)l * SEQ * DH,
                                  kvc + (size_t)(NL + l) * SEQ * DH,
                                  cmask, pos, o);
    k_gemv<NH * DH><<<HID / 16, 256, 0, stream>>>(Wo + (size_t)l * HID * NH * DH, o, t1);
    k_add_rms<<<1, 256, 0, stream>>>(h, t1, ln_pa + l * HID, HID);

    k_rms<<<1, 256, 0, stream>>>(h, ln_pf + l * HID, xb, HID);
    k_gemv<HID><<<FF / 16, 256, 0, stream>>>(Wg + (size_t)l * FF * HID, xb, gate);
    k_gemv<HID><<<FF / 16, 256, 0, stream>>>(Wu + (size_t)l * FF * HID, xb, up);
    k_gelu_mul<<<FF / 256, 256, 0, stream>>>(gate, up, gate, FF);
    k_gemv<FF><<<HID / 16, 256, 0, stream>>>(Wd + (size_t)l * HID * FF, gate, t1);
    k_add_rms<<<1, 256, 0, stream>>>(h, t1, ln_ff + l * HID, HID);
  }

  k_rms<<<1, 256, 0, stream>>>(h, nw, xb, HID);
  k_gemv<HID><<<VOCAB / 16, 256, 0, stream>>>(lmh, xb, logits);
  k_argmax<<<1, 1024, 0, stream>>>(logits, (float*)d_out);
}